// FourierPointProcess_17051020165137
// MI455X (gfx1250) — compile-verified
//
#include <hip/hip_runtime.h>
#include <math.h>

typedef __attribute__((ext_vector_type(2))) float v2f;
typedef __attribute__((ext_vector_type(8))) float v8f;

#define NF_ 512
#define NS_ 64
#define FS_ 256
#define B_  32
#define S_  1024
#define D_  3
#define MU_ 10.0f
// -MU * T * (2*pi)^(D-1) = -10 * 10 * (2pi)^2
#define LL2_ (-3947.8417604357435f)

#ifndef __has_builtin
#define __has_builtin(x) 0
#endif

// ---------------------------------------------------------------- utilities
__global__ void zero_f32_kernel(float* __restrict__ p, int n) {
    int i = blockIdx.x * blockDim.x + threadIdx.x;
    if (i < n) p[i] = 0.0f;
}

// ------------------------------------------------- MLP stage 1: H = tanh(noise@W1+b1)
__global__ void mlp1_kernel(const float* __restrict__ noise,
                            const float* __restrict__ W1,
                            const float* __restrict__ b1,
                            float* __restrict__ H) {
    int idx = blockIdx.x * blockDim.x + threadIdx.x;
    if (idx >= NF_ * 100) return;
    int nf = idx / 100;
    int j  = idx - nf * 100;
    float acc = b1[j];
#pragma unroll
    for (int k = 0; k < NS_; ++k)
        acc = fmaf(noise[nf * NS_ + k], W1[k * 100 + j], acc);
    H[idx] = tanhf(acc);
}

// ---------------------- MLP stage 2 + projection: Womg4[n][0..2] = (tanh(H@W2+b2) @ W)^T
__global__ void womg_kernel(const float* __restrict__ H,
                            const float* __restrict__ W2,
                            const float* __restrict__ b2,
                            const float* __restrict__ W,
                            float* __restrict__ Womg4) {
    int t = blockIdx.x * blockDim.x + threadIdx.x;
    if (t >= NF_) return;
    float hr[100];
#pragma unroll
    for (int k = 0; k < 100; ++k) hr[k] = H[t * 100 + k];
    float w0 = 0.0f, w1 = 0.0f, w2 = 0.0f;
    for (int f = 0; f < FS_; ++f) {          // W2 column f is uniform -> scalar loads
        float acc = b2[f];
#pragma unroll
        for (int k = 0; k < 100; ++k) acc = fmaf(hr[k], W2[k * FS_ + f], acc);
        float fv = tanhf(acc);
        w0 = fmaf(fv, W[f * 3 + 0], w0);
        w1 = fmaf(fv, W[f * 3 + 1], w1);
        w2 = fmaf(fv, W[f * 3 + 2], w2);
    }
    Womg4[t * 4 + 0] = w0;
    Womg4[t * 4 + 1] = w1;
    Womg4[t * 4 + 2] = w2;
    Womg4[t * 4 + 3] = 0.0f;   // K=4 pad for WMMA
}

// ---------------------------------------------------------------- main fused kernel
// grid (B_, 4): block handles batch b and a 128-wide Fourier-column chunk.
// 128 threads = 4 waves; each wave owns 32 columns as two 16-col WMMA groups.
// Sequentially walks s in tiles of 16:
//   theta tile (16x16) via V_WMMA_F32_16X16X4_F32  (A = X rows, K=4 padded)
//   cos/sin, exclusive scan (intra-lane prefix + shfl_xor(16) half handoff),
//   per-row reduce over 16 lanes, atomicAdd into ksum[b][s].
__global__ void __launch_bounds__(128)
hawkes_main_kernel(const float* __restrict__ X,
                   const float* __restrict__ Womg4,
                   float* __restrict__ ksum) {
    const int b      = blockIdx.x;
    const int nchunk = blockIdx.y;
    const int wave   = threadIdx.x >> 5;
    const int lane   = threadIdx.x & 31;
    const int half   = (lane & 16) ? 1 : 0;
    const int l15    = lane & 15;
    const int n0     = nchunk * 128 + wave * 32;
    const float ymask = half ? 0.0f : 1.0f;   // kills a.y for K=3 pad lanes, no branch

    // B fragments for the two 16-column groups (constant across s).
    // B is 4x16: lanes 0-15 hold K=0 (.x) / K=1 (.y); lanes 16-31 hold K=2 / K=3(=0).
    v2f bfr[2];
#pragma unroll
    for (int g = 0; g < 2; ++g) {
        int n    = n0 + g * 16 + l15;
        bfr[g].x = Womg4[n * 4 + (half ? 2 : 0)];
        bfr[g].y = half ? 0.0f : Womg4[n * 4 + 1];
    }

    float runC[2] = {0.0f, 0.0f};
    float runS[2] = {0.0f, 0.0f};
    const float* xr = X + ((size_t)b * S_ + l15) * D_;  // my A-row pointer, advances 16 rows/tile

    for (int s0 = 0; s0 < S_; s0 += 16, xr += 16 * D_) {
        // A fragment: A[M=l15][K]; lanes 0-15 carry K=0/1, lanes 16-31 K=2/3(=0).
        // Load all 3 components unconditionally (contiguous -> b96), select with cndmask.
        float x0 = xr[0];
        float x1 = xr[1];
        float x2 = xr[2];
        v2f a;
        a.x = half ? x2 : x0;
        a.y = x1 * ymask;

        float kacc[8] = {0, 0, 0, 0, 0, 0, 0, 0};
#pragma unroll
        for (int g = 0; g < 2; ++g) {
            v8f dtile;
#if __has_builtin(__builtin_amdgcn_wmma_f32_16x16x4_f32)
            v8f cz = {};
            dtile = __builtin_amdgcn_wmma_f32_16x16x4_f32(
                false, a, false, bfr[g], (short)0, cz, false, false);
#else
            // VALU fallback producing the same D layout:
            // lane holds column n0+g*16+l15, rows r + half*8
#pragma unroll
            for (int r = 0; r < 8; ++r) {
                const float* xx = X + ((size_t)b * S_ + s0 + r + half * 8) * D_;
                const float* wv = Womg4 + (n0 + g * 16 + l15) * 4;
                dtile[r] = fmaf(xx[0], wv[0], fmaf(xx[1], wv[1], xx[2] * wv[2]));
            }
            (void)a;
#endif
            // cos/sin + local (8-row) totals
            float cv[8], sv[8], sc = 0.0f, ss = 0.0f;
#pragma unroll
            for (int r = 0; r < 8; ++r) {
                float th = dtile[r];
                cv[r] = __cosf(th);
                sv[r] = __sinf(th);
                sc += cv[r];
                ss += sv[r];
            }
            // half-to-half handoff of prefix totals (column lives in lane pair l, l^16)
            float psc = __shfl_xor(sc, 16, 32);
            float pss = __shfl_xor(ss, 16, 32);
            float pc  = runC[g] + (half ? psc : 0.0f);   // exclusive base for my rows
            float ps  = runS[g] + (half ? pss : 0.0f);
#pragma unroll
            for (int r = 0; r < 8; ++r) {
                kacc[r] = fmaf(cv[r], pc, fmaf(sv[r], ps, kacc[r]));
                pc += cv[r];
                ps += sv[r];
            }
            runC[g] += sc + psc;   // both half-lanes keep the full column cumsum
            runS[g] += ss + pss;
        }
        // reduce over the 16 lanes of each half (rows 0-7 lower, 8-15 upper)
#pragma unroll
        for (int r = 0; r < 8; ++r) {
            float v = kacc[r];
            v += __shfl_xor(v, 1, 32);
            v += __shfl_xor(v, 2, 32);
            v += __shfl_xor(v, 4, 32);
            v += __shfl_xor(v, 8, 32);
            if (l15 == 0)
                atomicAdd(&ksum[b * S_ + s0 + r + half * 8], v);
        }
    }
}

// ---------------------------------------------------------------- finalize
__global__ void finalize_kernel(const float* __restrict__ X,
                                const float* __restrict__ ksum,
                                const float* __restrict__ alpha,
                                float* __restrict__ out) {
    const int SP1 = S_ + 1;
    int idx = blockIdx.x * blockDim.x + threadIdx.x;
    if (idx >= B_ * SP1) return;
    int b = idx / SP1;
    int s = idx - b * SP1;
    float* lam_out = out;             // [B_, S_]
    float* ll_out  = out + B_ * S_;   // [B_, S_+1]
    if (s < S_) {
        float k   = ksum[b * S_ + s] * (1.0f / (float)NF_);
        float lam = fmaf(alpha[0], k, MU_);
        lam_out[b * S_ + s] = lam;
        float mask = (X[(size_t)(b * S_ + s) * D_] > 0.0f) ? 1.0f : 0.0f;
        ll_out[idx] = __logf(lam) * mask + LL2_;
    } else {
        ll_out[idx] = LL2_;
    }
}

// ---------------------------------------------------------------- launcher
extern "C" void kernel_launch(void* const* d_in, const int* in_sizes, int n_in,
                              void* d_out, int out_size, void* d_ws, size_t ws_size,
                              hipStream_t stream) {
    const float* X     = (const float*)d_in[0];
    const float* noise = (const float*)d_in[1];
    const float* W1    = (const float*)d_in[2];
    const float* b1    = (const float*)d_in[3];
    const float* W2    = (const float*)d_in[4];
    const float* b2    = (const float*)d_in[5];
    const float* W     = (const float*)d_in[6];
    const float* alpha = (const float*)d_in[7];

    float* ws    = (float*)d_ws;
    float* H     = ws;                       // 512*100 = 51200 floats
    float* Womg4 = ws + 51200;               // 512*4   =  2048 floats
    float* ksum  = ws + 51200 + 2048;        // 32*1024 = 32768 floats

    hipLaunchKernelGGL(zero_f32_kernel, dim3((B_ * S_ + 255) / 256), dim3(256), 0,
                       stream, ksum, B_ * S_);
    hipLaunchKernelGGL(mlp1_kernel, dim3((NF_ * 100 + 255) / 256), dim3(256), 0,
                       stream, noise, W1, b1, H);
    hipLaunchKernelGGL(womg_kernel, dim3((NF_ + 255) / 256), dim3(256), 0,
                       stream, H, W2, b2, W, Womg4);
    hipLaunchKernelGGL(hawkes_main_kernel, dim3(B_, 4), dim3(128), 0,
                       stream, X, Womg4, ksum);
    hipLaunchKernelGGL(finalize_kernel, dim3((B_ * (S_ + 1) + 255) / 256), dim3(256), 0,
                       stream, X, ksum, alpha, (float*)d_out);
}